// DenoisingNetwork_56710748176537
// MI455X (gfx1250) — compile-verified
//
#include <hip/hip_runtime.h>
#include <hip/hip_bf16.h>

typedef __attribute__((ext_vector_type(2))) float v2f;
typedef __attribute__((ext_vector_type(8))) float v8f;

#define B_ 2
#define N_ 16384
#define C_ 64
#define K_ 16
#define O_ 128
#define M_ (B_ * N_ * K_)      // 524288 rows through the MLP
#define KA1 68                 // 64 feat + 3 pos, padded to multiple of 4

// ---------------------------------------------------------------------------
// 1) xyz_approx = mean over K of position_condition ; store packed float4
// ---------------------------------------------------------------------------
__global__ __launch_bounds__(256) void mean_xyz_kernel(
    const float* __restrict__ pos, float4* __restrict__ xyz) {
  int i = blockIdx.x * 256 + threadIdx.x;          // over B*N
  const float* p = pos + (size_t)i * K_ * 3;
  float sx = 0.f, sy = 0.f, sz = 0.f;
#pragma unroll
  for (int j = 0; j < K_; ++j) {
    sx += p[3 * j + 0];
    sy += p[3 * j + 1];
    sz += p[3 * j + 2];
  }
  const float s = 1.0f / (float)K_;
  xyz[i] = make_float4(sx * s, sy * s, sz * s, 0.f);
}

// ---------------------------------------------------------------------------
// 2) brute-force KNN, k=16, per batch. 256 queries per block, LDS point tiles,
//    register-resident sorted top-16 with guarded insert.
// ---------------------------------------------------------------------------
__global__ __launch_bounds__(256) void knn_kernel(
    const float4* __restrict__ xyz, int* __restrict__ nbr) {
  __shared__ float4 pts[256];
  const int tid = threadIdx.x;
  const int q = blockIdx.x * 256 + tid;            // global over B*N
  const int b = q >> 14;                           // q / N_
  const float4 qp = xyz[q];
  const float4* base = xyz + (size_t)b * N_;

  float bd[K_];
  int bi[K_];
#pragma unroll
  for (int j = 0; j < K_; ++j) { bd[j] = 3.4e38f; bi[j] = 0; }

  for (int tile = 0; tile < N_ / 256; ++tile) {
    __syncthreads();
    pts[tid] = base[tile * 256 + tid];
    __syncthreads();
#pragma unroll 4
    for (int t = 0; t < 256; ++t) {
      float4 p = pts[t];
      float dx = qp.x - p.x, dy = qp.y - p.y, dz = qp.z - p.z;
      float d = dx * dx + dy * dy + dz * dz;
      if (d < bd[K_ - 1]) {                        // rare: ~0.7% of candidates
        float dj = d;
        int ij = tile * 256 + t;
#pragma unroll
        for (int j = 0; j < K_; ++j) {             // sorted bubble-insert
          bool lt = dj < bd[j];
          float tf = bd[j]; int ti = bi[j];
          bd[j] = lt ? dj : bd[j];
          bi[j] = lt ? ij : bi[j];
          dj = lt ? tf : dj;
          ij = lt ? ti : ij;
        }
      }
    }
  }
#pragma unroll
  for (int j = 0; j < K_; ++j) nbr[(size_t)q * K_ + j] = bi[j];
}

// ---------------------------------------------------------------------------
// 3) build X[M x 68] = [features[b, nbr[row], :64] | pos[row,:3] | 0]
//    (flat nbr index == MLP row index)
// ---------------------------------------------------------------------------
__global__ __launch_bounds__(256) void build_input_kernel(
    const float* __restrict__ feat, const float* __restrict__ pos,
    const int* __restrict__ nbr, float* __restrict__ X) {
  const long total = (long)M_ * 17;                // 17 float4 segs per row
  const long stride = (long)gridDim.x * 256;
  for (long idx = (long)blockIdx.x * 256 + threadIdx.x; idx < total;
       idx += stride) {
    long row = idx / 17;
    int seg = (int)(idx - row * 17);
    float4 v;
    if (seg < 16) {
      int b = (int)(row >> 18);                    // row / (N*K), N*K = 2^18
      int j = nbr[row];
      const float4* f = (const float4*)(feat + ((size_t)(b * N_ + j) * C_));
      v = f[seg];
    } else {
      const float* pp = pos + row * 3;
      v = make_float4(pp[0], pp[1], pp[2], 0.f);
    }
    *(float4*)(X + row * KA1 + seg * 4) = v;
  }
}

// ---------------------------------------------------------------------------
// 4) fp32 WMMA GEMM: H[M,128] = A[M,lda(kaPad)] * W[kaReal,128] + bias
//    64 rows x 128 cols per 256-thread block (8 waves, wave32).
//    W staged in LDS K-pair interleaved: ldsW[(k/2)*256 + 2n + (k&1)], so each
//    lane's B fragment {W[k][n], W[k+1][n]} is one contiguous ds_load_b64.
//    maxpool==1: layer-3 epilogue, max over the 16-row tile -> out[B*N,128]
// ---------------------------------------------------------------------------
__global__ __launch_bounds__(256) void gemm_wmma_kernel(
    const float* __restrict__ A, const float* __restrict__ W,
    const float* __restrict__ bias, float* __restrict__ out,
    int kaReal, int kaPad, int lda, int maxpool) {
  extern __shared__ float ldsW[];                  // kaPad * 128 floats
  const int tid = threadIdx.x;

  for (int idx = tid; idx < kaPad * 128; idx += 256) {
    int k = idx >> 7, n = idx & 127;
    float v = (k < kaReal) ? W[k * 128 + n] : 0.0f;
    ldsW[((k >> 1) << 8) + (n << 1) + (k & 1)] = v;  // pair-interleaved
  }
  __syncthreads();

  const int wave = tid >> 5;
  const int lane = tid & 31;
  const int laneLo = lane & 15;
  const int laneHi = lane >> 4;                    // 0|1 : K pair select
  const int wm = wave >> 1;                        // 0..3 -> 16-row slab
  const int wn = wave & 1;                         // 0..1 -> 64-col slab
  const long m0 = (long)blockIdx.x * 64 + (long)wm * 16;
  const int nbase = wn * 64;

  v8f acc[4] = {};                                 // 16 x 64 per wave
  const float* arow = A + (m0 + laneLo) * (long)lda;
  // per-lane LDS base: pair row laneHi, column (nbase+laneLo)
  const float* wbase = ldsW + (laneHi << 8) + ((nbase + laneLo) << 1);

  for (int k0 = 0; k0 < kaPad; k0 += 4) {
    const int krow = k0 + (laneHi << 1);
    // A 16x4 fragment: lanes 0-15 -> K0,K1 ; lanes 16-31 -> K2,K3
    v2f a = *(const v2f*)(arow + krow);
    const float* wp = wbase + ((k0 >> 1) << 8);    // advance 2 pair-rows
#pragma unroll
    for (int t = 0; t < 4; ++t) {
      v2f bf = *(const v2f*)(wp + (t << 5));       // one ds_load_b64
      acc[t] = __builtin_amdgcn_wmma_f32_16x16x4_f32(
          false, a, false, bf, (short)0, acc[t], false, false);
    }
  }

  if (!maxpool) {
#pragma unroll
    for (int t = 0; t < 4; ++t) {
      const int col = nbase + t * 16 + laneLo;
      const float bc = bias[col];
      const long rbase = m0 + (long)laneHi * 8;
#pragma unroll
      for (int r = 0; r < 8; ++r)
        out[(rbase + r) * 128 + col] = acc[t][r] + bc;
    }
  } else {
    // one 16-row tile == one (b,n) neighbor group -> maxpool in registers
#pragma unroll
    for (int t = 0; t < 4; ++t) {
      const int col = nbase + t * 16 + laneLo;
      const float bc = bias[col];
      float mx = -3.4e38f;
#pragma unroll
      for (int r = 0; r < 8; ++r) mx = fmaxf(mx, acc[t][r] + bc);
      mx = fmaxf(mx, __shfl_xor(mx, 16));          // combine row halves
      if (laneHi == 0) out[(m0 >> 4) * 128 + col] = mx;
    }
  }
}

// ---------------------------------------------------------------------------
// 5) BatchNorm pass 1: per-channel sum / sumsq (channel constant per thread)
// ---------------------------------------------------------------------------
__global__ __launch_bounds__(256) void bn_stats_kernel(
    const float* __restrict__ H, float* __restrict__ stats) {
  __shared__ float ssum[O_], ssq[O_];
  const int tid = threadIdx.x;
  if (tid < O_) { ssum[tid] = 0.f; ssq[tid] = 0.f; }
  __syncthreads();
  const long total = (long)M_ * O_;
  const long stride = (long)gridDim.x * blockDim.x;  // multiple of 128
  float s = 0.f, q = 0.f;
  for (long i = (long)blockIdx.x * blockDim.x + tid; i < total; i += stride) {
    float v = H[i];
    s += v;
    q += v * v;
  }
  const int c = tid & (O_ - 1);
  atomicAdd(&ssum[c], s);
  atomicAdd(&ssq[c], q);
  __syncthreads();
  if (tid < O_) {
    atomicAdd(&stats[tid], ssum[tid]);
    atomicAdd(&stats[O_ + tid], ssq[tid]);
  }
}

__global__ void bn_finalize_kernel(const float* __restrict__ stats,
                                   const float* __restrict__ g,
                                   const float* __restrict__ be,
                                   float* __restrict__ coef) {
  const int c = threadIdx.x;                       // 128 threads
  const float inv = 1.0f / (float)M_;
  float mean = stats[c] * inv;
  float var = stats[O_ + c] * inv - mean * mean;   // biased variance
  float sc = g[c] * rsqrtf(var + 1e-5f);
  coef[c] = sc;
  coef[O_ + c] = be[c] - mean * sc;
}

__global__ __launch_bounds__(256) void bn_apply_kernel(
    float* __restrict__ H, const float* __restrict__ coef) {
  const long total = (long)M_ * O_ / 4;
  const int c0 = (threadIdx.x * 4) & (O_ - 1);     // constant per thread
  const float4 sc = *(const float4*)(coef + c0);
  const float4 sh = *(const float4*)(coef + O_ + c0);
  const long stride = (long)gridDim.x * blockDim.x;
  for (long i = (long)blockIdx.x * blockDim.x + threadIdx.x; i < total;
       i += stride) {
    float4 v = *(float4*)(H + i * 4);
    v.x = fmaxf(0.f, v.x * sc.x + sh.x);
    v.y = fmaxf(0.f, v.y * sc.y + sh.y);
    v.z = fmaxf(0.f, v.z * sc.z + sh.z);
    v.w = fmaxf(0.f, v.w * sc.w + sh.w);
    *(float4*)(H + i * 4) = v;
  }
}

// ---------------------------------------------------------------------------
extern "C" void kernel_launch(void* const* d_in, const int* in_sizes, int n_in,
                              void* d_out, int out_size, void* d_ws,
                              size_t ws_size, hipStream_t stream) {
  const float* features = (const float*)d_in[0];   // [B,N,64]
  const float* poscond  = (const float*)d_in[1];   // [B,N,16,3]
  const float* W1 = (const float*)d_in[2];         // [67,128]
  const float* b1 = (const float*)d_in[3];
  const float* g1 = (const float*)d_in[4];
  const float* be1 = (const float*)d_in[5];
  const float* W2 = (const float*)d_in[6];         // [128,128]
  const float* b2 = (const float*)d_in[7];
  const float* g2 = (const float*)d_in[8];
  const float* be2 = (const float*)d_in[9];
  const float* W3 = (const float*)d_in[10];        // [128,128]
  const float* b3 = (const float*)d_in[11];
  float* out = (float*)d_out;                      // [B,N,128]

  float* ws = (float*)d_ws;
  float4* xyz = (float4*)ws;                                   // B*N float4
  int* nbr = (int*)(ws + (size_t)B_ * N_ * 4);                 // M ints
  float* X  = ws + (size_t)B_ * N_ * 4 + M_;                   // M*68
  float* H1 = X + (size_t)M_ * KA1;                            // M*128
  float* H2 = H1 + (size_t)M_ * O_;                            // M*128
  float* stats = H2 + (size_t)M_ * O_;                         // 256
  float* coef = stats + 256;                                   // 256

  const int BN = B_ * N_;

  mean_xyz_kernel<<<BN / 256, 256, 0, stream>>>(poscond, xyz);
  knn_kernel<<<BN / 256, 256, 0, stream>>>(xyz, nbr);
  build_input_kernel<<<4096, 256, 0, stream>>>(features, poscond, nbr, X);

  // layer 1: X[M,68] @ W1[67->68,128] + b1  ->  H1
  gemm_wmma_kernel<<<M_ / 64, 256, KA1 * 128 * sizeof(float), stream>>>(
      X, W1, b1, H1, 67, KA1, KA1, 0);
  hipMemsetAsync(stats, 0, 256 * sizeof(float), stream);
  bn_stats_kernel<<<1024, 256, 0, stream>>>(H1, stats);
  bn_finalize_kernel<<<1, 128, 0, stream>>>(stats, g1, be1, coef);
  bn_apply_kernel<<<4096, 256, 0, stream>>>(H1, coef);

  // layer 2: H1 @ W2 + b2 -> H2
  gemm_wmma_kernel<<<M_ / 64, 256, O_ * 128 * sizeof(float), stream>>>(
      H1, W2, b2, H2, O_, O_, O_, 0);
  hipMemsetAsync(stats, 0, 256 * sizeof(float), stream);
  bn_stats_kernel<<<1024, 256, 0, stream>>>(H2, stats);
  bn_finalize_kernel<<<1, 128, 0, stream>>>(stats, g2, be2, coef);
  bn_apply_kernel<<<4096, 256, 0, stream>>>(H2, coef);

  // layer 3: H2 @ W3 + b3, maxpool over 16 neighbors -> out[B*N,128]
  gemm_wmma_kernel<<<M_ / 64, 256, O_ * 128 * sizeof(float), stream>>>(
      H2, W3, b3, out, O_, O_, O_, 1);
}